// ScoerModel_90898687852781
// MI455X (gfx1250) — compile-verified
//
#include <hip/hip_runtime.h>

#define BB 512
#define SEQN 50
#define HD 256
#define MEMN 8
#define VOC 40001
#define NITEMS (VOC - 1)
#define NEGF (-1.0e12f)
#define SCALEF 0.0625f /* 1/sqrt(256) */

typedef __attribute__((ext_vector_type(2))) float v2f;
typedef __attribute__((ext_vector_type(8))) float v8f;

__device__ __forceinline__ v8f wmma_f32(v2f a, v2f b, v8f c) {
  // V_WMMA_F32_16X16X4_F32 : D = A(16x4) * B(4x16) + C(16x16), fp32 throughout
  return __builtin_amdgcn_wmma_f32_16x16x4_f32(false, a, false, b, (short)0, c,
                                               false, false);
}

// ---------------------------------------------------------------------------
// Sinusoid positional table TAB[50][256] (double-precision, matches reference)
// ---------------------------------------------------------------------------
__global__ void tab_kernel(float* __restrict__ tab) {
  int idx = blockIdx.x * blockDim.x + threadIdx.x;
  if (idx >= SEQN * HD) return;
  int t = idx / HD, d = idx % HD;
  double expo = (double)(2 * (d / 2)) / (double)HD;
  double ang = (double)t / pow(10000.0, expo);
  tab[idx] = (d & 1) ? (float)cos(ang) : (float)sin(ang);
}

__global__ void zerof_kernel(float* __restrict__ p, size_t n) {
  size_t i = (size_t)blockIdx.x * blockDim.x + threadIdx.x;
  if (i < n) p[i] = 0.f;
}

// last[b] = sum(mask[b]) - 1 ; sr[b] = seq_emb_1[b, last[b]]
__global__ void last_sr_kernel(const int* __restrict__ mask,
                               const float* __restrict__ seq1,
                               int* __restrict__ last, float* __restrict__ sr) {
  int b = blockIdx.x, tid = threadIdx.x;
  __shared__ int sl;
  if (tid == 0) {
    int s = 0;
    for (int t = 0; t < SEQN; ++t) s += mask[b * SEQN + t];
    sl = s - 1;
    last[b] = sl;
  }
  __syncthreads();
  sr[(size_t)b * HD + tid] = seq1[((size_t)b * SEQN + sl) * HD + tid];
}

// ---------------------------------------------------------------------------
// Fused GRU timestep: gi = x_t @ wih^T, gh = h @ whh^T, gates, h update.
// One wave per 16x16 (batch x hidden) tile; 6 WMMA accumulators (r/z/n x gi/gh).
// ---------------------------------------------------------------------------
__global__ void gru_step_kernel(const float* __restrict__ x,
                                const float* __restrict__ hprev,
                                float* __restrict__ hnew, float* __restrict__ y,
                                const float* __restrict__ wih,
                                const float* __restrict__ whh,
                                const float* __restrict__ bih,
                                const float* __restrict__ bhh,
                                const int* __restrict__ mask, int t,
                                int mask_out) {
  int wave = (blockIdx.x << 3) + (threadIdx.x >> 5);
  if (wave >= (BB / 16) * (HD / 16)) return;
  int mt = wave / (HD / 16);
  int nt = wave % (HD / 16);
  int l = threadIdx.x & 31;
  int lm = l & 15;
  int kk = (l >> 4) << 1;
  int marow = mt * 16 + lm;
  int ncol = nt * 16 + lm;

  const float* xr = x + (size_t)marow * (SEQN * HD) + (size_t)t * HD;
  const float* hr = hprev + (size_t)marow * HD;
  const float* wir = wih + (size_t)(0 * HD + ncol) * HD;
  const float* wiz = wih + (size_t)(1 * HD + ncol) * HD;
  const float* win = wih + (size_t)(2 * HD + ncol) * HD;
  const float* whr = whh + (size_t)(0 * HD + ncol) * HD;
  const float* whz = whh + (size_t)(1 * HD + ncol) * HD;
  const float* whn = whh + (size_t)(2 * HD + ncol) * HD;

  v8f air = {0, 0, 0, 0, 0, 0, 0, 0}, aiz = air, ain = air;
  v8f ahr = air, ahz = air, ahn = air;
  for (int k = 0; k < HD; k += 4) {
    v2f ax, ah, bm;
    ax.x = xr[k + kk]; ax.y = xr[k + kk + 1];
    ah.x = hr[k + kk]; ah.y = hr[k + kk + 1];
    bm.x = wir[k + kk]; bm.y = wir[k + kk + 1]; air = wmma_f32(ax, bm, air);
    bm.x = wiz[k + kk]; bm.y = wiz[k + kk + 1]; aiz = wmma_f32(ax, bm, aiz);
    bm.x = win[k + kk]; bm.y = win[k + kk + 1]; ain = wmma_f32(ax, bm, ain);
    bm.x = whr[k + kk]; bm.y = whr[k + kk + 1]; ahr = wmma_f32(ah, bm, ahr);
    bm.x = whz[k + kk]; bm.y = whz[k + kk + 1]; ahz = wmma_f32(ah, bm, ahz);
    bm.x = whn[k + kk]; bm.y = whn[k + kk + 1]; ahn = wmma_f32(ah, bm, ahn);
  }
  int nl = nt * 16 + lm;
  for (int i = 0; i < 8; ++i) {
    int m = mt * 16 + ((l < 16) ? i : i + 8);
    float ir = air[i] + bih[nl];
    float hrv = ahr[i] + bhh[nl];
    float iz = aiz[i] + bih[HD + nl];
    float hz = ahz[i] + bhh[HD + nl];
    float in_ = ain[i] + bih[2 * HD + nl];
    float hn = ahn[i] + bhh[2 * HD + nl];
    float r = 1.f / (1.f + expf(-(ir + hrv)));
    float z = 1.f / (1.f + expf(-(iz + hz)));
    float nn = tanhf(in_ + r * hn);
    float hp = hprev[(size_t)m * HD + nl];
    float h2 = (1.f - z) * nn + z * hp;
    hnew[(size_t)m * HD + nl] = h2;
    float yo = h2;
    if (mask_out) yo *= (float)mask[m * SEQN + t];
    y[((size_t)m * SEQN + t) * HD + nl] = yo;
  }
}

// ---------------------------------------------------------------------------
// Generic WMMA GEMM: C[M,Nc] = A[M,K] @ W  (one wave per 16x16 tile)
// flags: 1=W stored [Nc,K] (x@W.T), 2=scale rows by mask[m], 4=add TAB[|t-last|],
//        8=accumulate into C, 16=ReLU
// ---------------------------------------------------------------------------
__global__ void gemm16_kernel(const float* __restrict__ A, int lda,
                              const float* __restrict__ W, float* __restrict__ C,
                              int ldc, int M, int Nc, int K,
                              const float* __restrict__ bias,
                              const float* __restrict__ bias2,
                              const int* __restrict__ mask,
                              const int* __restrict__ last,
                              const float* __restrict__ tab, int flags) {
  int tiles_n = Nc >> 4;
  int wave = (blockIdx.x << 3) + (threadIdx.x >> 5);
  if (wave >= (M >> 4) * tiles_n) return;
  int mt = wave / tiles_n, nt = wave % tiles_n;
  int l = threadIdx.x & 31;
  int lm = l & 15;
  int kk = (l >> 4) << 1;
  const float* arow = A + (size_t)(mt * 16 + lm) * lda;
  int nb = nt * 16 + lm;
  v8f acc = {0, 0, 0, 0, 0, 0, 0, 0};
  if (flags & 1) {
    const float* wrow = W + (size_t)nb * K;
    for (int k = 0; k < K; k += 4) {
      v2f a, b;
      a.x = arow[k + kk]; a.y = arow[k + kk + 1];
      b.x = wrow[k + kk]; b.y = wrow[k + kk + 1];
      acc = wmma_f32(a, b, acc);
    }
  } else {
    for (int k = 0; k < K; k += 4) {
      v2f a, b;
      a.x = arow[k + kk]; a.y = arow[k + kk + 1];
      b.x = W[(size_t)(k + kk) * Nc + nb];
      b.y = W[(size_t)(k + kk + 1) * Nc + nb];
      acc = wmma_f32(a, b, acc);
    }
  }
  for (int i = 0; i < 8; ++i) {
    int m = mt * 16 + ((l < 16) ? i : i + 8);
    int n = nt * 16 + lm;
    float v = acc[i];
    if (flags & 2) v *= (float)mask[m];
    if (flags & 4) {
      int b = m / SEQN, tq = m % SEQN;
      int d = tq - last[b];
      if (d < 0) d = -d;
      v += tab[(size_t)d * HD + n];
    }
    if (bias) v += bias[n];
    if (bias2) v += bias2[n];
    size_t ci = (size_t)m * ldc + n;
    if (flags & 8) v += C[ci];
    if (flags & 16) v = fmaxf(v, 0.f);
    C[ci] = v;
  }
}

// ---------------------------------------------------------------------------
// Attention core: one block per (b, t). use_mem=1 prepends 8 always-allowed
// memory slots. Causal + pad masking, softmax, weighted value sum, * mask.
// ---------------------------------------------------------------------------
__global__ void attn_core_kernel(const float* __restrict__ Q,
                                 const float* __restrict__ Km,
                                 const float* __restrict__ Vm,
                                 const float* __restrict__ memk,
                                 const float* __restrict__ memv,
                                 const int* __restrict__ mask,
                                 float* __restrict__ out, int use_mem) {
  int b = blockIdx.x / SEQN, t = blockIdx.x % SEQN;
  int tid = threadIdx.x;
  __shared__ float qs[HD];
  __shared__ float sc[MEMN + SEQN];
  qs[tid] = Q[((size_t)b * SEQN + t) * HD + tid];
  __syncthreads();
  int nk = use_mem ? (MEMN + SEQN) : SEQN;
  if (tid < nk) {
    const float* kp;
    bool allow;
    if (use_mem && tid < MEMN) {
      kp = memk + (size_t)tid * HD;
      allow = true;
    } else {
      int j = use_mem ? tid - MEMN : tid;
      kp = Km + ((size_t)b * SEQN + j) * HD;
      allow = (j <= t) && (mask[b * SEQN + j] != 0);
    }
    float s = 0.f;
    for (int h = 0; h < HD; ++h) s += qs[h] * kp[h];
    sc[tid] = allow ? s * SCALEF : NEGF;
  }
  __syncthreads();
  if (tid == 0) {
    float mx = -3.0e38f;
    for (int j = 0; j < nk; ++j) mx = fmaxf(mx, sc[j]);
    float sum = 0.f;
    for (int j = 0; j < nk; ++j) {
      float e = expf(sc[j] - mx);
      sc[j] = e;
      sum += e;
    }
    float inv = 1.f / sum;
    for (int j = 0; j < nk; ++j) sc[j] *= inv;
  }
  __syncthreads();
  float mf = (float)mask[b * SEQN + t];
  float accv = 0.f;
  for (int j = 0; j < nk; ++j) {
    const float* vp = (use_mem && j < MEMN)
                          ? (memv + (size_t)j * HD)
                          : (Vm + ((size_t)b * SEQN + (use_mem ? j - MEMN : j)) * HD);
    accv += sc[j] * vp[tid];
  }
  out[((size_t)b * SEQN + t) * HD + tid] = accv * mf;
}

// alpha logits -> masked softmax over t -> sess pooling; build [sess, sr] concat
__global__ void alpha_sess_kernel(const float* __restrict__ ATT,
                                  const float* __restrict__ q11,
                                  const float* __restrict__ W3,
                                  const int* __restrict__ mask,
                                  const float* __restrict__ seq1,
                                  const float* __restrict__ sr,
                                  float* __restrict__ cat) {
  int b = blockIdx.x, tid = threadIdx.x;
  __shared__ float red[HD];
  __shared__ float lg[SEQN];
  red[tid] = q11[(size_t)b * HD + tid] * W3[HD + tid];
  __syncthreads();
  for (int s = HD / 2; s > 0; s >>= 1) {
    if (tid < s) red[tid] += red[tid + s];
    __syncthreads();
  }
  float l2 = red[0];
  if (tid < SEQN) {
    const float* ar = ATT + ((size_t)b * SEQN + tid) * HD;
    float s = 0.f;
    for (int h = 0; h < HD; ++h) s += ar[h] * W3[h];
    lg[tid] = s + l2;
  }
  __syncthreads();
  if (tid == 0) {
    float mx = -3.0e38f;
    for (int t = 0; t < SEQN; ++t) {
      float x = mask[b * SEQN + t] ? lg[t] : NEGF;
      lg[t] = x;
      mx = fmaxf(mx, x);
    }
    float sum = 0.f;
    for (int t = 0; t < SEQN; ++t) {
      float e = expf(lg[t] - mx);
      lg[t] = e;
      sum += e;
    }
    float inv = 1.f / sum;
    for (int t = 0; t < SEQN; ++t) lg[t] *= inv;
  }
  __syncthreads();
  float v = 0.f;
  for (int t = 0; t < SEQN; ++t)
    v += lg[t] * (float)mask[b * SEQN + t] * seq1[((size_t)b * SEQN + t) * HD + tid];
  cat[(size_t)b * (2 * HD) + tid] = v;
  cat[(size_t)b * (2 * HD) + HD + tid] = sr[(size_t)b * HD + tid];
}

__global__ void beta_kernel(const float* __restrict__ U,
                            const float* __restrict__ Wt2,
                            float* __restrict__ beta) {
  int b = blockIdx.x, tid = threadIdx.x;
  __shared__ float r0[HD], r1[HD];
  float u = U[(size_t)b * HD + tid];
  r0[tid] = u * Wt2[tid * 2];
  r1[tid] = u * Wt2[tid * 2 + 1];
  __syncthreads();
  for (int s = HD / 2; s > 0; s >>= 1) {
    if (tid < s) {
      r0[tid] += r0[tid + s];
      r1[tid] += r1[tid + s];
    }
    __syncthreads();
  }
  if (tid == 0) {
    float l0 = r0[0], l1 = r1[0];
    float m = fmaxf(l0, l1);
    float e0 = expf(l0 - m), e1 = expf(l1 - m), inv = 1.f / (e0 + e1);
    beta[b * 2] = e0 * inv;
    beta[b * 2 + 1] = e1 * inv;
  }
}

__global__ void boundary_kernel(const float* __restrict__ beta,
                                float* __restrict__ outp) {
  __shared__ float red[BB];
  int tid = threadIdx.x;
  red[tid] = fabsf(beta[tid * 2] - beta[tid * 2 + 1]);
  __syncthreads();
  for (int s = BB / 2; s > 0; s >>= 1) {
    if (tid < s) red[tid] += red[tid + s];
    __syncthreads();
  }
  if (tid == 0) outp[(size_t)BB * NITEMS] = red[0] / (float)BB;
}

// Online (max,sum) for the two masked softmaxes over the 40000-item vocab.
__global__ void score_stats_kernel(const float* __restrict__ score,
                                   const int* __restrict__ fl,
                                   float* __restrict__ stats) {
  int b = blockIdx.x, tid = threadIdx.x;
  float mi = -3.0e38f, si = 0.f, me = -3.0e38f, se = 0.f;
  const float* srow = score + (size_t)b * NITEMS;
  const int* frow = fl + (size_t)b * NITEMS;
  for (int i = tid; i < NITEMS; i += blockDim.x) {
    float s = srow[i];
    int f = frow[i];
    float xin = f ? s : NEGF;
    float xex = f ? NEGF : s;
    if (xin > mi) { si = si * expf(mi - xin) + 1.f; mi = xin; }
    else si += expf(xin - mi);
    if (xex > me) { se = se * expf(me - xex) + 1.f; me = xex; }
    else se += expf(xex - me);
  }
  __shared__ float smi[256], ssi[256], sme[256], sse[256];
  smi[tid] = mi; ssi[tid] = si; sme[tid] = me; sse[tid] = se;
  __syncthreads();
  for (int s = 128; s > 0; s >>= 1) {
    if (tid < s) {
      float m1 = smi[tid], m2 = smi[tid + s], Mx = fmaxf(m1, m2);
      ssi[tid] = ssi[tid] * expf(m1 - Mx) + ssi[tid + s] * expf(m2 - Mx);
      smi[tid] = Mx;
      m1 = sme[tid]; m2 = sme[tid + s]; Mx = fmaxf(m1, m2);
      sse[tid] = sse[tid] * expf(m1 - Mx) + sse[tid + s] * expf(m2 - Mx);
      sme[tid] = Mx;
    }
    __syncthreads();
  }
  if (tid == 0) {
    stats[b * 4 + 0] = smi[0];
    stats[b * 4 + 1] = ssi[0];
    stats[b * 4 + 2] = sme[0];
    stats[b * 4 + 3] = sse[0];
  }
}

__global__ void final_kernel(const float* __restrict__ score,
                             const int* __restrict__ fl,
                             const float* __restrict__ stats,
                             const float* __restrict__ beta,
                             float* __restrict__ outp) {
  size_t idx = (size_t)blockIdx.x * blockDim.x + threadIdx.x;
  if (idx >= (size_t)BB * NITEMS) return;
  int b = (int)(idx / NITEMS);
  float s = score[idx];
  int f = fl[idx];
  float xin = f ? s : NEGF;
  float xex = f ? NEGF : s;
  float pin = expf(xin - stats[b * 4 + 0]) / stats[b * 4 + 1];
  float pex = expf(xex - stats[b * 4 + 2]) / stats[b * 4 + 3];
  outp[idx] = pin * beta[b * 2] + pex * beta[b * 2 + 1];
}

// ---------------------------------------------------------------------------
extern "C" void kernel_launch(void* const* d_in, const int* in_sizes, int n_in,
                              void* d_out, int out_size, void* d_ws,
                              size_t ws_size, hipStream_t stream) {
  const float* seq1 = (const float*)d_in[0];
  const float* seq2 = (const float*)d_in[1];
  const int* mask = (const int*)d_in[2];
  const int* item_flag = (const int*)d_in[3];
  const float* wih0 = (const float*)d_in[4];
  const float* whh0 = (const float*)d_in[5];
  const float* bih0 = (const float*)d_in[6];
  const float* bhh0 = (const float*)d_in[7];
  const float* wih1 = (const float*)d_in[8];
  const float* whh1 = (const float*)d_in[9];
  const float* bih1 = (const float*)d_in[10];
  const float* bhh1 = (const float*)d_in[11];
  const float* Wq1 = (const float*)d_in[12];
  const float* Wk1 = (const float*)d_in[13];
  const float* Wv1 = (const float*)d_in[14];
  /* d_in[15] = Wq2 : unused by reference */
  const float* Wk2 = (const float*)d_in[16];
  const float* Wv2 = (const float*)d_in[17];
  const float* Wffn1 = (const float*)d_in[18];
  const float* bffn1 = (const float*)d_in[19];
  const float* Wffn2 = (const float*)d_in[20];
  const float* bffn2 = (const float*)d_in[21];
  const float* Wone1 = (const float*)d_in[22];
  const float* bone1 = (const float*)d_in[23];
  const float* W3 = (const float*)d_in[24];
  const float* Wtrans = (const float*)d_in[25];
  const float* btrans = (const float*)d_in[26];
  const float* Wtail1 = (const float*)d_in[27];
  const float* btail1 = (const float*)d_in[28];
  const float* Wtail2 = (const float*)d_in[29];
  const float* memk = (const float*)d_in[30];
  const float* memv = (const float*)d_in[31];
  const float* emb1 = (const float*)d_in[32];
  float* out = (float*)d_out;
  float* ws = (float*)d_ws;

  const size_t S = (size_t)BB * SEQN * HD; // 6,553,600 floats
  float* S0 = ws;
  float* S1 = ws + S;
  float* S2 = ws + 2 * S;
  float* S3 = ws + 3 * S;
  float* S4 = ws + 4 * S;
  float* S5 = ws + 5 * S;
  float* sm = ws + 6 * S;
  float* tab = sm; sm += SEQN * HD;
  float* h0 = sm; sm += (size_t)BB * HD;
  float* h1 = sm; sm += (size_t)BB * HD;
  float* sr = sm; sm += (size_t)BB * HD;
  int* last = (int*)sm; sm += 512;
  float* q11 = sm; sm += (size_t)BB * HD;
  float* cat = sm; sm += (size_t)BB * 2 * HD;
  float* sessF = sm; sm += (size_t)BB * HD;
  float* U = sm; sm += (size_t)BB * HD;
  float* beta = sm; sm += 2 * BB;
  float* stats = sm; sm += 4 * BB;
  float* score = S0; // S0..S5 are dead by the time score is computed

  auto gemm = [&](const float* A, int lda, const float* W, float* C, int ldc,
                  int M, int Nc, int K, const float* bias, const float* bias2,
                  int flags) {
    int tiles = (M >> 4) * (Nc >> 4);
    gemm16_kernel<<<(tiles + 7) / 8, 256, 0, stream>>>(
        A, lda, W, C, ldc, M, Nc, K, bias, bias2, mask, last, tab, flags);
  };

  tab_kernel<<<(SEQN * HD + 255) / 256, 256, 0, stream>>>(tab);
  last_sr_kernel<<<BB, HD, 0, stream>>>(mask, seq1, last, sr);

  // --- two GRU layers, ping-pong hidden state, one launch per timestep ---
  const int gru_blocks = ((BB / 16) * (HD / 16) + 7) / 8;
  zerof_kernel<<<((BB * HD) + 255) / 256, 256, 0, stream>>>(h0, (size_t)BB * HD);
  for (int t = 0; t < SEQN; ++t) {
    const float* hin = (t & 1) ? h1 : h0;
    float* hout = (t & 1) ? h0 : h1;
    gru_step_kernel<<<gru_blocks, 256, 0, stream>>>(seq2, hin, hout, S0, wih0,
                                                    whh0, bih0, bhh0, mask, t, 0);
  }
  zerof_kernel<<<((BB * HD) + 255) / 256, 256, 0, stream>>>(h0, (size_t)BB * HD);
  for (int t = 0; t < SEQN; ++t) {
    const float* hin = (t & 1) ? h1 : h0;
    float* hout = (t & 1) ? h0 : h1;
    gru_step_kernel<<<gru_blocks, 256, 0, stream>>>(S0, hin, hout, S1, wih1,
                                                    whh1, bih1, bhh1, mask, t, 1);
  }

  const int Mqkv = BB * SEQN;
  // q1/k1/v1 = (seq1*mask) @ W + TAB[|t-last|]
  gemm(seq1, HD, Wq1, S2, HD, Mqkv, HD, HD, nullptr, nullptr, 2 | 4);
  gemm(seq1, HD, Wk1, S3, HD, Mqkv, HD, HD, nullptr, nullptr, 2 | 4);
  gemm(seq1, HD, Wv1, S4, HD, Mqkv, HD, HD, nullptr, nullptr, 2 | 4);
  attn_core_kernel<<<BB * SEQN, HD, 0, stream>>>(S2, S3, S4, nullptr, nullptr,
                                                 mask, S5, 0);
  gemm(S5, HD, Wffn1, S3, HD, Mqkv, HD, HD, bffn1, nullptr, 0); // ATT := att1
  gemm(S1, HD, Wk2, S4, HD, Mqkv, HD, HD, nullptr, nullptr, 0); // k2
  gemm(S1, HD, Wv2, S0, HD, Mqkv, HD, HD, nullptr, nullptr, 0); // v2
  attn_core_kernel<<<BB * SEQN, HD, 0, stream>>>(S2, S4, S0, memk, memv, mask,
                                                 S5, 1);
  gemm(S5, HD, Wffn2, S3, HD, Mqkv, HD, HD, bffn2, nullptr, 8 | 16); // relu(att1+att2)

  gemm(sr, HD, Wone1, q11, HD, BB, HD, HD, bone1, /*TAB[0]=*/tab, 0);
  alpha_sess_kernel<<<BB, HD, 0, stream>>>(S3, q11, W3, mask, seq1, sr, cat);
  gemm(cat, 2 * HD, Wtrans, sessF, HD, BB, HD, 2 * HD, btrans, nullptr, 0);
  gemm(sessF, HD, Wtail1, U, HD, BB, HD, HD, btail1, nullptr, 16);
  beta_kernel<<<BB, HD, 0, stream>>>(U, Wtail2, beta);
  boundary_kernel<<<1, BB, 0, stream>>>(beta, out);

  // score = sessF @ emb1[1:].T  (W stored [Nc,K] -> WT mode)
  gemm(sessF, HD, emb1 + HD, score, NITEMS, BB, NITEMS, HD, nullptr, nullptr, 1);
  score_stats_kernel<<<BB, 256, 0, stream>>>(score, item_flag, stats);
  size_t tot = (size_t)BB * NITEMS;
  final_kernel<<<(int)((tot + 255) / 256), 256, 0, stream>>>(score, item_flag,
                                                             stats, beta, out);
}